// Resample2d_5669356835728
// MI455X (gfx1250) — compile-verified
//
#include <hip/hip_runtime.h>

// Bilinear flow-warp (Resample2d) for MI455X (gfx1250, wave32).
//
// Memory-bound gather kernel: ~478 MB of HBM traffic, ~0.5 GFLOP.
// Floor at 23.3 TB/s is ~20.5 us. Strategy:
//  - batch index uniform per block -> all gathers AND stores are saddr-form
//    (SGPR base + 32-bit lane offset, scale_offset), immediate plane offsets
//  - 4 corner offsets + 4 weights computed once, amortized over 64 channels
//  - non-temporal stores keep the 235 MB output stream out of the 192 MB L2
//    so each batch's 29 MB image slab stays resident for corner reuse
//  - unroll 8 -> 32 outstanding gathers/wave; prefetch one full unrolled
//    chunk (8 planes = 3.67 MB) ahead via global_prefetch_b8

__device__ __forceinline__ int iclamp(int v, int lo, int hi) {
    return min(max(v, lo), hi);
}

template <int C, int H, int W>
__global__ __launch_bounds__(256)
void resample2d_warp_kernel(const float* __restrict__ img,   // [B,C,H,W]
                            const float* __restrict__ flow,  // [B,2,H,W]
                            float* __restrict__ out)         // [B,C,H,W]
{
    constexpr int HW  = H * W;          // 114688 = 448 blocks * 256 threads
    constexpr int BPB = HW / 256;       // blocks per batch (448)
    constexpr int CHW = C * HW;
    constexpr int UNROLL = 8;

    // Batch index is uniform per block -> base pointers stay in SGPRs.
    const int b   = blockIdx.x / BPB;
    const int pib = (blockIdx.x % BPB) * 256 + threadIdx.x;   // pixel-in-batch
    const int y   = pib / W;
    const int x   = pib - y * W;

    // Flow components (coalesced b32 loads, read once -> keep default RT).
    const float* fb = flow + (size_t)b * 2 * HW;
    const float dx = fb[pib];
    const float dy = fb[HW + pib];

    // Bilinear coordinates / weights (computed once, reused for 64 channels).
    const float xf  = (float)x + dx;
    const float yf  = (float)y + dy;
    const float x0f = floorf(xf);
    const float y0f = floorf(yf);
    const float a   = xf - x0f;       // alpha (unclamped, matches reference)
    const float bt  = yf - y0f;       // beta

    const int x0i = (int)x0f;
    const int y0i = (int)y0f;
    const int x0 = iclamp(x0i,     0, W - 1);
    const int x1 = iclamp(x0i + 1, 0, W - 1);
    const int y0 = iclamp(y0i,     0, H - 1);
    const int y1 = iclamp(y0i + 1, 0, H - 1);

    const float w00 = (1.0f - a) * (1.0f - bt);
    const float w01 = a          * (1.0f - bt);
    const float w10 = (1.0f - a) * bt;
    const float w11 = a          * bt;

    // Element offsets inside one channel plane (fit easily in 32 bits).
    const int o00 = y0 * W + x0;
    const int o01 = y0 * W + x1;
    const int o10 = y1 * W + x0;
    const int o11 = y1 * W + x1;

    const float* p  = img + (size_t)b * CHW;   // SGPR base for gathers
    float*       ob = out + (size_t)b * CHW;   // SGPR base for stores

    #pragma unroll UNROLL
    for (int c = 0; c < C; ++c) {
        // Prefetch one unrolled chunk ahead (8 planes = 3.67 MB): the 8
        // iterations of a chunk are clause-batched, so a 1-plane distance
        // would land inside the same issue group. Speculative prefetch past
        // the buffer end (last chunk) is silently dropped per the CDNA5 ISA.
        __builtin_prefetch((const void*)(p + UNROLL * HW + o00), 0, 3);

        const float g00 = p[o00];
        const float g01 = p[o01];
        const float g10 = p[o10];
        const float g11 = p[o11];

        const float v = fmaf(w00, g00,
                        fmaf(w01, g01,
                        fmaf(w10, g10, w11 * g11)));

        // saddr-form NT store: SGPR base + pib lane offset, plane stride in
        // the 24-bit immediate. Output is streamed/never re-read -> TH=NT.
        __builtin_nontemporal_store(v, ob + pib);

        p  += HW;
        ob += HW;
    }
}

extern "C" void kernel_launch(void* const* d_in, const int* in_sizes, int n_in,
                              void* d_out, int out_size, void* d_ws, size_t ws_size,
                              hipStream_t stream) {
    constexpr int B = 8, C = 64, H = 256, W = 448;

    const float* img  = (const float*)d_in[0];  // (8, 64, 256, 448) fp32
    const float* flow = (const float*)d_in[1];  // (8, 2, 256, 448) fp32
    float*       out  = (float*)d_out;          // (8, 64, 256, 448) fp32

    const int blocks = B * ((H * W) / 256);     // 3584 blocks x 8 wave32 waves
    resample2d_warp_kernel<C, H, W><<<blocks, 256, 0, stream>>>(img, flow, out);
}